// CustomGATv2Layer_40132174414162
// MI455X (gfx1250) — compile-verified
//
#include <hip/hip_runtime.h>
#include <math.h>

#define N_NODES   50000
#define DEG       16
#define IN_FEATS  128
#define HEADS     8
#define OUT_FEATS 16
#define HF        (HEADS * OUT_FEATS)   // 128
#define KSTEPS    (IN_FEATS / 4)        // 32 WMMA K-steps

typedef __attribute__((ext_vector_type(2))) float v2f;
typedef __attribute__((ext_vector_type(4))) float v4f;
typedef __attribute__((ext_vector_type(8))) float v8f;

// ---------------------------------------------------------------------------
// Kernel 1: hp = h @ W_fc^T via V_WMMA_F32_16X16X4_F32.
// W_fc (64KB) is staged once per block into LDS in *fragment order*:
//   Wlds[(t*8 + j)*64 + lane*2 + q] = B-frag element for K-step t, col-tile j.
// Fragment layout (wave32): lane<16 holds (col=16j+lane, K=4t,4t+1) in v0/v1;
// lane>=16 holds (col=16j+lane-16, K=4t+2,4t+3).  A strip is preloaded into
// 32 v2f registers so the main loop is pure ds_load_b64 + v_wmma.
// Fused epilogue: s[n,head] = dot(hp[n, head*16:+16], w_attn).
// ---------------------------------------------------------------------------
__global__ void gat_gemm_proj_kernel(const float* __restrict__ h,
                                     const float* __restrict__ W,      // [128 out][128 in]
                                     const float* __restrict__ w_attn, // [16]
                                     float* __restrict__ hp,           // [N][128]
                                     float* __restrict__ s)            // [N][8]
{
    __shared__ float Wlds[KSTEPS * 8 * 64];          // 16384 floats = 64 KB exactly

    const int tid  = threadIdx.x;
    const int lane = tid & 31;

    // ---- Cooperative staging of W into fragment-order LDS (all 256 threads) ----
    // Global element (row, col): row = out-feature (0..127), col = K (0..127).
    //   t = col>>2, j = row>>4, lane = (row&15) + 16*((col>>1)&1), q = col&1
    //   -> lds[(t*8+j)*64 + 2*(row&15) + 32*((col>>1)&1) + (col&1)]
    #pragma unroll
    for (int it = 0; it < 16; ++it) {
        const int g    = tid + it * 256;             // 4096 v4-chunks total
        const int row  = g >> 5;                     // 32 v4-chunks per row
        const int c4   = g & 31;                     // chunk -> cols 4*c4 .. 4*c4+3
        const v4f w    = *(const v4f*)(W + (size_t)row * IN_FEATS + c4 * 4);
        const int base = (c4 * 8 + (row >> 4)) * 64 + ((row & 15) << 1);
        Wlds[base + 0]  = w.x;                       // (kh=0, q=0)
        Wlds[base + 1]  = w.y;                       // (kh=0, q=1)
        Wlds[base + 32] = w.z;                       // (kh=2, q=0) -> lane+16
        Wlds[base + 33] = w.w;                       // (kh=2, q=1)
    }
    __syncthreads();

    const int wid = (blockIdx.x * blockDim.x + tid) >> 5;
    if (wid < N_NODES / 16) {                        // wave-uniform guard (after barrier)
        const int ml = lane & 15;
        const int kh = (lane >> 4) << 1;             // 0 or 2
        const int m0 = wid * 16;

        // ---- Preload entire A strip for this lane: 32 x v2f (64 VGPRs) ----
        const float* __restrict__ arow = h + (size_t)(m0 + ml) * IN_FEATS + kh;
        v2f areg[KSTEPS];
        #pragma unroll
        for (int t = 0; t < KSTEPS; ++t)
            areg[t] = *(const v2f*)(arow + 4 * t);

        v8f acc[8];
        #pragma unroll
        for (int j = 0; j < 8; ++j) acc[j] = (v8f){0.f,0.f,0.f,0.f,0.f,0.f,0.f,0.f};

        // ---- Main loop: conflict-free ds_load_b64 + v_wmma, fully unrolled ----
        #pragma unroll
        for (int t = 0; t < KSTEPS; ++t) {
            v2f bf[8];
            #pragma unroll
            for (int j = 0; j < 8; ++j)
                bf[j] = *(const v2f*)(&Wlds[(t * 8 + j) * 64 + (lane << 1)]);
            #pragma unroll
            for (int j = 0; j < 8; ++j)
                acc[j] = __builtin_amdgcn_wmma_f32_16x16x4_f32(
                    false, areg[t], false, bf[j], (short)0, acc[j], false, false);
        }

        // ---- Epilogue: store hp tile + s = acc * w_attn reduced over 16 cols ----
        const float watt = w_attn[ml];
        #pragma unroll
        for (int j = 0; j < 8; ++j) {
            #pragma unroll
            for (int r = 0; r < 8; ++r) {
                // C layout: VGPR r -> row m0+r (lanes 0-15), m0+r+8 (lanes 16-31)
                const int row = m0 + r + ((lane >> 4) << 3);
                const float v = acc[j][r];
                hp[(size_t)row * HF + 16 * j + ml] = v;
                float p = v * watt;                  // 16-lane half-wave sum
                p += __shfl_xor(p, 1);
                p += __shfl_xor(p, 2);
                p += __shfl_xor(p, 4);
                p += __shfl_xor(p, 8);
                if (ml == 0) s[(size_t)row * HEADS + j] = p;
            }
        }
    }
}

// ---------------------------------------------------------------------------
// Kernel 2: per-destination-node softmax attention + weighted aggregation.
// dst = repeat(arange(N), 16) -> edges of node n are [n*16, n*16+16).
// One wave per node. Phase 1: lane = (edge k = lane%16, 4 heads per half).
// Phase 2 (via LDS): lane = 4 consecutive output features (one b128 gather/edge).
// ---------------------------------------------------------------------------
__global__ void gat_attn_agg_kernel(const float* __restrict__ hp,   // [N][128]
                                    const float* __restrict__ s,    // [N][8]
                                    const float* __restrict__ ew,   // [E]
                                    const int*   __restrict__ src,  // [E]
                                    const float* __restrict__ bias, // [128]
                                    float* __restrict__ out)        // [N][128]
{
    __shared__ float alpha_lds[8][DEG * HEADS];  // per-wave slots
    __shared__ int   src_lds[8][DEG];

    const int n     = (blockIdx.x * blockDim.x + threadIdx.x) >> 5;  // exact: 6250*8 waves
    const int lane  = threadIdx.x & 31;
    const int wslot = (threadIdx.x >> 5);

    const int k  = lane & 15;                    // edge within segment
    const int hb = (lane >> 4) << 2;             // head base: 0 or 4

    const int   sidx = src[n * DEG + k];
    const float lw   = log1pf(ew[n * DEG + k]);
    if (lane < 16) src_lds[wslot][k] = sidx;

    #pragma unroll
    for (int q = 0; q < 4; ++q) {
        const int hq = hb + q;
        float x = s[(size_t)sidx * HEADS + hq] + s[(size_t)n * HEADS + hq] + lw;
        x = (x > 0.f) ? x : 0.01f * x;           // leaky_relu, slope 0.01
        // max over the 16 edges (within 16-lane half)
        float m = x;
        m = fmaxf(m, __shfl_xor(m, 1));
        m = fmaxf(m, __shfl_xor(m, 2));
        m = fmaxf(m, __shfl_xor(m, 4));
        m = fmaxf(m, __shfl_xor(m, 8));
        float ex = expf(x - m);
        float d = ex;
        d += __shfl_xor(d, 1);
        d += __shfl_xor(d, 2);
        d += __shfl_xor(d, 4);
        d += __shfl_xor(d, 8);
        alpha_lds[wslot][k * HEADS + hq] = ex / d;
    }

    __syncthreads();   // all 256 threads reach this (no partial blocks)

    // Aggregation: lane owns features c..c+3 (all within head c/16).
    const int c  = lane << 2;
    const int hc = lane >> 2;                    // head of features c..c+3
    float a0 = 0.f, a1 = 0.f, a2 = 0.f, a3 = 0.f;
    #pragma unroll
    for (int kk = 0; kk < DEG; ++kk) {
        const int   sk = src_lds[wslot][kk];
        const float al = alpha_lds[wslot][kk * HEADS + hc];
        const v4f   v  = *(const v4f*)(hp + (size_t)sk * HF + c);
        a0 += al * v.x;
        a1 += al * v.y;
        a2 += al * v.z;
        a3 += al * v.w;
    }
    const v4f b4 = *(const v4f*)(bias + c);
    v4f r;
    r.x = a0 + b4.x;
    r.y = a1 + b4.y;
    r.z = a2 + b4.z;
    r.w = a3 + b4.w;
    *(v4f*)(out + (size_t)n * HF + c) = r;
}

// ---------------------------------------------------------------------------
extern "C" void kernel_launch(void* const* d_in, const int* in_sizes, int n_in,
                              void* d_out, int out_size, void* d_ws, size_t ws_size,
                              hipStream_t stream) {
    const float* h       = (const float*)d_in[0];   // [N,128]
    const float* ew      = (const float*)d_in[1];   // [E]
    const int*   src     = (const int*)d_in[2];     // [E]
    // d_in[3] = dst: structurally repeat(arange(N),16) -> contiguous segments, unused
    const float* W_fc    = (const float*)d_in[4];   // [128,128]
    const float* w_attn  = (const float*)d_in[5];   // [16]
    const float* bias    = (const float*)d_in[6];   // [128]
    float*       out     = (float*)d_out;           // [N,128]

    float* hp = (float*)d_ws;                                 // N*128 f32 = 25.6 MB
    float* s  = hp + (size_t)N_NODES * HF;                    // N*8   f32 =  1.6 MB

    // Kernel 1: 3125 waves (one per 16-row strip), 8 waves per block
    {
        const int waves  = N_NODES / 16;                      // 3125
        const int blocks = (waves * 32 + 255) / 256;          // 391
        gat_gemm_proj_kernel<<<blocks, 256, 0, stream>>>(h, W_fc, w_attn, hp, s);
    }
    // Kernel 2: one wave per node, 50000 waves = exactly 6250 blocks of 256
    {
        const int blocks = (N_NODES * 32) / 256;              // 6250
        gat_attn_agg_kernel<<<blocks, 256, 0, stream>>>(hp, s, ew, src, bias, out);
    }
}